// ScaledDotProductAttention_30064771072760
// MI455X (gfx1250) — compile-verified
//
#include <hip/hip_runtime.h>

// Relation-aware attention for MI455X (gfx1250, wave32, WMMA).
// B=2, H=8, L=512, D=64. Memory-bound: adj_k/adj_v (134 MB each) read once.

constexpr int B = 2, H = 8, L = 512, D = 64;
constexpr float INV_TEMP = 0.125f; // 1/sqrt(D) = 1/8

typedef __attribute__((ext_vector_type(2))) float v2f;
typedef __attribute__((ext_vector_type(8))) float v8f;

__device__ inline v8f wmma_f32_k4(v2f a, v2f b, v8f c) {
  // D = A(16x4,f32) * B(4x16,f32) + C(16x16,f32)
  return __builtin_amdgcn_wmma_f32_16x16x4_f32(
      /*neg_a=*/false, a, /*neg_b=*/false, b,
      /*c_mod=*/(short)0, c, /*reuse_a=*/false, /*reuse_b=*/false);
}

// ---------------------------------------------------------------------------
// Kernel 1: S[b,h,i,j] = (q/8) . k   (per-head GEMM 512x512x64)
// one wave per 16x16 output tile; waves = B*H*32*32 = 16384
// ---------------------------------------------------------------------------
__global__ void qk_base_kernel(const float* __restrict__ q,
                               const float* __restrict__ k,
                               float* __restrict__ attn) {
  const int lane = threadIdx.x & 31;
  const int wid  = blockIdx.x * 8 + (threadIdx.x >> 5);
  const int jt = wid & 31;
  const int it = (wid >> 5) & 31;
  const int bh = wid >> 10;           // 0..15

  const float* qb = q + (size_t)bh * L * D;
  const float* kb = k + (size_t)bh * L * D;
  float* Sb = attn + (size_t)bh * L * L;

  const int m   = lane & 15;          // A row / B col within tile
  const int kb2 = (lane >> 4) * 2;    // K sub-pair base
  const int i0 = it * 16, j0 = jt * 16;

  const float* arow = qb + (size_t)(i0 + m) * D + kb2;
  const float* brow = kb + (size_t)(j0 + m) * D + kb2; // B[k][n] = K[j0+n][k]

  v8f c = {};
  #pragma unroll 4
  for (int kc = 0; kc < 16; ++kc) {
    v2f a = *(const v2f*)(arow + kc * 4);
    a[0] *= INV_TEMP; a[1] *= INV_TEMP;
    v2f bm = *(const v2f*)(brow + kc * 4);
    c = wmma_f32_k4(a, bm, c);
  }
  const int rbase = 8 * (lane >> 4);
  #pragma unroll
  for (int r = 0; r < 8; ++r)
    Sb[(size_t)(i0 + rbase + r) * L + j0 + m] = c[r];
}

// ---------------------------------------------------------------------------
// Kernel 2: S[b,:,i,j] += (q[b,:,i,:]/8) @ adj_k[b,i,:,:]^T
// per (b,i): 8x512x64 GEMM (8 head-rows padded to M=16); streams adj_k once.
// one wave per (b, i, j-tile); waves = B*L*32 = 32768
// ---------------------------------------------------------------------------
__global__ void rel_score_kernel(const float* __restrict__ q,
                                 const float* __restrict__ adj_k,
                                 float* __restrict__ attn) {
  const int lane = threadIdx.x & 31;
  const int wid  = blockIdx.x * 8 + (threadIdx.x >> 5);
  const int jt = wid & 31;
  const int i  = (wid >> 5) & (L - 1);
  const int b  = wid >> 14;

  const int m   = lane & 15;          // head (A row) and j-col (B col)
  const int kb2 = (lane >> 4) * 2;
  const int j0 = jt * 16;

  const bool avalid = m < H;
  const float* arow =
      q + (((size_t)(b * H + (avalid ? m : 0)) * L + i) * D) + kb2;
  const float* adj = adj_k + ((size_t)(b * L + i) * L + j0) * D; // rows j0..j0+15

  v8f c = {};
  #pragma unroll 4
  for (int kc = 0; kc < 16; ++kc) {
    v2f a = {0.f, 0.f};
    if (avalid) {
      v2f t = *(const v2f*)(arow + kc * 4);
      a[0] = t[0] * INV_TEMP; a[1] = t[1] * INV_TEMP;
    }
    v2f bm = *(const v2f*)(adj + (size_t)m * D + kc * 4 + kb2);
    c = wmma_f32_k4(a, bm, c);
  }
  #pragma unroll
  for (int r = 0; r < 8; ++r) {
    const int h = r + 8 * (lane >> 4);
    if (h < H) {
      float* p = attn + (((size_t)(b * H + h) * L + i) * L) + j0 + m;
      *p += c[r];
    }
  }
}

// ---------------------------------------------------------------------------
// Kernel 3: mask + softmax over j, in place. One wave per (b,h,i) row.
// waves = B*H*L = 8192; each lane owns 16 of the 512 columns.
// ---------------------------------------------------------------------------
__global__ void softmax_kernel(const int* __restrict__ mask,
                               float* __restrict__ attn) {
  const int lane = threadIdx.x & 31;
  const int row  = blockIdx.x * 8 + (threadIdx.x >> 5); // (b*H+h)*L + i
  const int b = row >> 12;                              // / (H*L)

  float* S = attn + (size_t)row * L;
  const int* mk = mask + b * L;

  float vals[16];
  float mx = -1e30f;
  #pragma unroll
  for (int t = 0; t < 16; ++t) {
    const int j = lane + t * 32;
    float s = S[j];
    s = (mk[j] == 0) ? -10000.0f : s;
    vals[t] = s;
    mx = fmaxf(mx, s);
  }
  #pragma unroll
  for (int off = 16; off > 0; off >>= 1)
    mx = fmaxf(mx, __shfl_xor(mx, off, 32));

  float sum = 0.f;
  #pragma unroll
  for (int t = 0; t < 16; ++t) {
    const float e = __expf(vals[t] - mx);
    vals[t] = e;
    sum += e;
  }
  #pragma unroll
  for (int off = 16; off > 0; off >>= 1)
    sum += __shfl_xor(sum, off, 32);

  const float inv = 1.0f / sum;
  #pragma unroll
  for (int t = 0; t < 16; ++t)
    S[lane + t * 32] = vals[t] * inv;
}

// ---------------------------------------------------------------------------
// Kernel 4: out[b,h,i,:] = attn[b,h,i,:] @ v[b,h,:,:]  (512x64x512 per head)
// one wave per 16x16 output tile; waves = B*H*32*4 = 2048
// ---------------------------------------------------------------------------
__global__ void av_base_kernel(const float* __restrict__ attn,
                               const float* __restrict__ v,
                               float* __restrict__ out) {
  const int lane = threadIdx.x & 31;
  const int wid  = blockIdx.x * 8 + (threadIdx.x >> 5);
  const int dt = wid & 3;
  const int it = (wid >> 2) & 31;
  const int bh = wid >> 7;

  const float* A = attn + (size_t)bh * L * L;
  const float* V = v + (size_t)bh * L * D;

  const int m   = lane & 15;
  const int kb2 = (lane >> 4) * 2;
  const int i0 = it * 16, d0 = dt * 16;

  const float* arow = A + (size_t)(i0 + m) * L + kb2;
  const float* bcol = V + (size_t)kb2 * D + d0 + m; // B[k][n] = V[k][d0+n]

  v8f c = {};
  #pragma unroll 4
  for (int kc = 0; kc < 128; ++kc) {
    v2f a = *(const v2f*)(arow + kc * 4);
    v2f bm;
    bm[0] = bcol[(size_t)(kc * 4) * D];
    bm[1] = bcol[(size_t)(kc * 4 + 1) * D];
    c = wmma_f32_k4(a, bm, c);
  }
  const int rbase = 8 * (lane >> 4);
  #pragma unroll
  for (int r = 0; r < 8; ++r)
    out[(size_t)bh * L * D + (size_t)(i0 + rbase + r) * D + d0 + m] = c[r];
}

// ---------------------------------------------------------------------------
// Kernel 5: out[b,:,i,:] += attn[b,:,i,:] @ adj_v[b,i,:,:]
// per (b,i): 8x64x512 GEMM (padded to M=16); streams adj_v once.
// one wave per (b, i, d-tile); waves = B*L*4 = 4096
// ---------------------------------------------------------------------------
__global__ void av_rel_kernel(const float* __restrict__ attn,
                              const float* __restrict__ adj_v,
                              float* __restrict__ out) {
  const int lane = threadIdx.x & 31;
  const int wid  = blockIdx.x * 8 + (threadIdx.x >> 5);
  const int dt = wid & 3;
  const int i  = (wid >> 2) & (L - 1);
  const int b  = wid >> 11;

  const int m   = lane & 15;          // head (A row) and d-col (B col)
  const int kb2 = (lane >> 4) * 2;
  const int d0 = dt * 16;

  const bool avalid = m < H;
  const float* arow =
      attn + (((size_t)(b * H + (avalid ? m : 0)) * L + i) * L) + kb2;
  const float* adv = adj_v + ((size_t)(b * L + i) * L) * D + d0 + m;

  v8f c = {};
  #pragma unroll 4
  for (int kc = 0; kc < 128; ++kc) {
    v2f a = {0.f, 0.f};
    if (avalid) a = *(const v2f*)(arow + kc * 4);
    v2f bm;
    bm[0] = adv[(size_t)(kc * 4 + kb2) * D];
    bm[1] = adv[(size_t)(kc * 4 + kb2 + 1) * D];
    c = wmma_f32_k4(a, bm, c);
  }
  #pragma unroll
  for (int r = 0; r < 8; ++r) {
    const int h = r + 8 * (lane >> 4);
    if (h < H) {
      float* p = out + (((size_t)(b * H + h) * L + i) * D) + d0 + m;
      *p += c[r];
    }
  }
}

// ---------------------------------------------------------------------------
extern "C" void kernel_launch(void* const* d_in, const int* in_sizes, int n_in,
                              void* d_out, int out_size, void* d_ws, size_t ws_size,
                              hipStream_t stream) {
  const float* q     = (const float*)d_in[0];
  const float* k     = (const float*)d_in[1];
  const float* v     = (const float*)d_in[2];
  const float* adj_k = (const float*)d_in[3];
  const float* adj_v = (const float*)d_in[4];
  const int*   mask  = (const int*)d_in[5];

  float* out  = (float*)d_out;                 // (B,H,L,D) = 524288 floats
  float* attn = out + (size_t)B * H * L * D;   // (B,H,L,L) = 4194304 floats

  // 256 threads/block = 8 wave32's
  qk_base_kernel  <<<2048, 256, 0, stream>>>(q, k, attn);
  rel_score_kernel<<<4096, 256, 0, stream>>>(q, adj_k, attn);
  softmax_kernel  <<<1024, 256, 0, stream>>>(mask, attn);
  av_base_kernel  <<< 256, 256, 0, stream>>>(attn, v, out);
  av_rel_kernel   <<< 512, 256, 0, stream>>>(attn, adj_v, out);
}